// Model_66073776882092
// MI455X (gfx1250) — compile-verified
//
#include <hip/hip_runtime.h>
#include <hip/hip_bf16.h>

typedef float v2f __attribute__((ext_vector_type(2)));
typedef float v8f __attribute__((ext_vector_type(8)));

__device__ inline void waveReduce2(float& a, float& b) {
#pragma unroll
  for (int off = 16; off > 0; off >>= 1) {
    a += __shfl_down(a, off, 32);
    b += __shfl_down(b, off, 32);
  }
}

// One wave = one 16-sample tile per iteration.
// A (16x4 f32 per step) = 16 gathered user rows; B (4x16) = 16 gathered movie rows
// as columns. 16 chained V_WMMA_F32_16X16X4_F32 over K=64 give D[i][j] = u_i . m_j;
// the diagonal is the 16 per-sample dots. Every gathered element is loaded exactly
// once per wave (lane n / lane n+16 split row n's K-range), and the reg-term squares
// are accumulated from the same loaded registers.
__global__ __launch_bounds__(256) void mf_loss_kernel(
    const int* __restrict__ u_ids, const int* __restrict__ m_ids,
    const float* __restrict__ ratings,
    const float* __restrict__ user_emb, const float* __restrict__ bias_user,
    const float* __restrict__ movie_emb, const float* __restrict__ bias_movie,
    const float* __restrict__ mean_p,
    float* __restrict__ ws, int N)
{
  const int lane = threadIdx.x & 31;
  const int wave = threadIdx.x >> 5;
  const int wavesPerBlock = blockDim.x >> 5;
  const int gwave = blockIdx.x * wavesPerBlock + wave;
  const int nwaves = gridDim.x * wavesPerBlock;
  const int ntiles = N >> 4;
  const float mean = *mean_p;

  const int n  = lane & 15;   // sample within tile (A row / B column)
  const int hi = lane >> 4;   // K sub-lane: 0 -> k%4 in {0,1}, 1 -> {2,3}

  float acc_loss = 0.f, acc_sq = 0.f;

  for (int tile = gwave; tile < ntiles; tile += nwaves) {
    const int i0 = tile << 4;
    const int uid = u_ids[i0 + n];
    const int mid = m_ids[i0 + n];
    const float* urow = user_emb  + ((size_t)uid << 6) + (hi << 1);
    const float* mrow = movie_emb + ((size_t)mid << 6) + (hi << 1);

    v8f c = {};
#pragma unroll
    for (int s = 0; s < 16; ++s) {
      // A layout 16x4 f32: lane n holds A[n][2*hi + v] in VGPR v
      v2f a = *(const v2f*)(urow + (s << 2));
      // B layout 4x16 f32: lane n holds B[2*hi + v][n] in VGPR v
      v2f b = *(const v2f*)(mrow + (s << 2));
      c = __builtin_amdgcn_wmma_f32_16x16x4_f32(false, a, false, b,
                                                (short)0, c, false, false);
      acc_sq += a.x * a.x + a.y * a.y + b.x * b.x + b.y * b.y;
    }

    // Diagonal owners: VGPR r -> sample r at lane r, sample 8+r at lane 24+r.
    // For every owner lane, its diagonal sample index equals n, so uid/mid/ratings
    // already loaded above are the right ones.
    const bool owner = (lane < 8) || (lane >= 24);
    if (owner) {
      const int r = (lane < 8) ? lane : (lane - 24);
      float dotv = c[0];
#pragma unroll
      for (int r2 = 1; r2 < 8; ++r2)
        if (r == r2) dotv = c[r2];
      const float e = ratings[i0 + n] -
                      (dotv + bias_user[uid] + bias_movie[mid] + mean);
      acc_loss += e * e;
    }
  }

  // Scalar tail for N % 16 (empty for N = 262144, kept for safety).
  const int remStart = ntiles << 4;
  if (blockIdx.x == 0 && wave == 0) {
    const int i = remStart + lane;
    if (i < N) {
      const int uid = u_ids[i], mid = m_ids[i];
      const float* urow = user_emb  + ((size_t)uid << 6);
      const float* mrow = movie_emb + ((size_t)mid << 6);
      float d = 0.f;
      for (int k = 0; k < 64; ++k) {
        const float uv = urow[k], mv = mrow[k];
        d += uv * mv;
        acc_sq += uv * uv + mv * mv;
      }
      const float e = ratings[i] - (d + bias_user[uid] + bias_movie[mid] + mean);
      acc_loss += e * e;
    }
  }

  // Wave -> block reduction, one (loss, sq) partial per block.
  waveReduce2(acc_loss, acc_sq);
  __shared__ float sl[8], ss[8];
  if (lane == 0) { sl[wave] = acc_loss; ss[wave] = acc_sq; }
  __syncthreads();
  if (wave == 0) {
    float l = (lane < wavesPerBlock) ? sl[lane] : 0.f;
    float s = (lane < wavesPerBlock) ? ss[lane] : 0.f;
#pragma unroll
    for (int off = 4; off > 0; off >>= 1) {
      l += __shfl_down(l, off, 32);
      s += __shfl_down(s, off, 32);
    }
    if (lane == 0) { ws[2 * blockIdx.x] = l; ws[2 * blockIdx.x + 1] = s; }
  }
}

// Single-block deterministic final reduction; applies loss + 2*lambda*sumsq.
__global__ __launch_bounds__(256) void mf_finalize_kernel(
    const float* __restrict__ ws, int nblocks,
    const float* __restrict__ lambda_p, float* __restrict__ out)
{
  float l = 0.f, s = 0.f;
  for (int i = threadIdx.x; i < nblocks; i += blockDim.x) {
    l += ws[2 * i];
    s += ws[2 * i + 1];
  }
  const int lane = threadIdx.x & 31;
  const int wave = threadIdx.x >> 5;
  waveReduce2(l, s);
  __shared__ float sl[8], ss[8];
  if (lane == 0) { sl[wave] = l; ss[wave] = s; }
  __syncthreads();
  if (threadIdx.x == 0) {
    float L = 0.f, S = 0.f;
#pragma unroll
    for (int w = 0; w < 8; ++w) { L += sl[w]; S += ss[w]; }
    out[0] = L + 2.0f * (*lambda_p) * S;
  }
}

extern "C" void kernel_launch(void* const* d_in, const int* in_sizes, int n_in,
                              void* d_out, int out_size, void* d_ws, size_t ws_size,
                              hipStream_t stream) {
  const int*   u_ids      = (const int*)d_in[0];
  const int*   m_ids      = (const int*)d_in[1];
  const float* ratings    = (const float*)d_in[2];
  const float* user_emb   = (const float*)d_in[3];
  const float* bias_user  = (const float*)d_in[4];
  const float* movie_emb  = (const float*)d_in[5];
  const float* bias_movie = (const float*)d_in[6];
  const float* mean_p     = (const float*)d_in[7];
  const float* lambda_p   = (const float*)d_in[8];
  const int N = in_sizes[0];

  int blocks = 1024;
  const int maxBlocksWs = (int)(ws_size / (2 * sizeof(float)));
  if (blocks > maxBlocksWs) blocks = maxBlocksWs;
  if (blocks < 1) blocks = 1;

  mf_loss_kernel<<<blocks, 256, 0, stream>>>(
      u_ids, m_ids, ratings, user_emb, bias_user, movie_emb, bias_movie,
      mean_p, (float*)d_ws, N);
  mf_finalize_kernel<<<1, 256, 0, stream>>>(
      (const float*)d_ws, blocks, lambda_p, (float*)d_out);
}